// SuperVoxelLoss_56169582297100
// MI455X (gfx1250) — compile-verified
//
#include <hip/hip_runtime.h>
#include <hip/hip_bf16.h>

// SuperVoxelLoss on MI455X (gfx1250, wave32).
// Pass 1: streaming pointwise loss + per-block fp32 partial sums (HBM-bound, ~6.7us floor).
// Pass 2: single-wave WMMA (V_WMMA_F32_16X16X4_F32) reduction of the partials -> mean.

typedef __attribute__((ext_vector_type(2))) float v2f;
typedef __attribute__((ext_vector_type(4))) float v4f;   // native vector: OK for nontemporal builtins
typedef __attribute__((ext_vector_type(8))) float v8f;

#define ALPHA 0.5f
#define BETA  0.5f

// total = bce * (1 + ALPHA*neg + BETA*pos); with ALPHA==BETA==0.5 and binary target,
// weight is 1.5 on threshold-disagreement, 1.0 otherwise. sigmoid(x)>0.5 <=> x>0.
__device__ __forceinline__ float loss_elem(float x, float t) {
    float ax   = __builtin_fabsf(x);
    float bce  = fmaxf(x, 0.0f) - x * t + log1pf(__expf(-ax));
    bool pred_fg = x > 0.0f;
    bool tgt_fg  = t > 0.5f;
    float w = (pred_fg != tgt_fg) ? 1.5f : 1.0f;
    return bce * w;
}

__global__ void svl_partial_kernel(const float* __restrict__ pred,
                                   const float* __restrict__ targ,
                                   float* __restrict__ partials,
                                   int n4, int n) {
    const int tid    = blockIdx.x * blockDim.x + threadIdx.x;
    const int stride = gridDim.x * blockDim.x;

    const v4f* __restrict__ p4 = (const v4f*)pred;
    const v4f* __restrict__ t4 = (const v4f*)targ;

    float acc = 0.0f;
    // Streaming 128-bit (global_load_b128) grid-stride loop; data is single-use,
    // so load non-temporally to avoid polluting WGP$/L2.
    for (int i = tid; i < n4; i += stride) {
        v4f xv = __builtin_nontemporal_load(&p4[i]);
        v4f tv = __builtin_nontemporal_load(&t4[i]);
        acc += loss_elem(xv.x, tv.x);
        acc += loss_elem(xv.y, tv.y);
        acc += loss_elem(xv.z, tv.z);
        acc += loss_elem(xv.w, tv.w);
    }
    // Scalar tail (N = 19,660,800 is divisible by 4, but stay general).
    for (int i = n4 * 4 + tid; i < n; i += stride) {
        acc += loss_elem(pred[i], targ[i]);
    }

    // wave32 butterfly reduction
    #pragma unroll
    for (int off = 16; off > 0; off >>= 1)
        acc += __shfl_xor(acc, off, 32);

    __shared__ float smem[32];
    const int lane = threadIdx.x & 31;
    const int wave = threadIdx.x >> 5;
    if (lane == 0) smem[wave] = acc;
    __syncthreads();

    if (threadIdx.x == 0) {
        const int nwaves = blockDim.x >> 5;
        float bsum = 0.0f;
        for (int w = 0; w < nwaves; ++w) bsum += smem[w];
        partials[blockIdx.x] = bsum;
    }
}

// Single block, single wave32. Reduces nPart (multiple of 64) fp32 partials
// using V_WMMA_F32_16X16X4_F32 with A == ones:
//   D[m][n] = sum_k A[m][k]*B[k][n] + C[m][n] = sum_k B[k][n] + C[m][n]
// Each issue folds 64 partials (2 B VGPRs) into the fp32 accumulator tile.
// Any packing of the 64 values into B is valid because we finally sum all 16
// columns (lanes 0..15 of C VGPR0 hold M=0, N=0..15).
__global__ void svl_final_wmma_kernel(const float* __restrict__ partials,
                                      float* __restrict__ out,
                                      int nPart, float invN) {
    const int lane = threadIdx.x;  // 0..31, EXEC all ones (uniform control flow)

    v2f a; a.x = 1.0f; a.y = 1.0f;   // A-matrix: 16x4 of ones
    v8f c = {};                      // C/D accumulator tile (16x16 f32)

    for (int base = 0; base < nPart; base += 64) {
        v2f b;
        b.x = partials[base + lane];
        b.y = partials[base + 32 + lane];
        // 8 args: (neg_a, A, neg_b, B, c_mod, C, reuse_a, reuse_b)
        c = __builtin_amdgcn_wmma_f32_16x16x4_f32(
                /*neg_a=*/false, a, /*neg_b=*/false, b,
                /*c_mod=*/(short)0, c, /*reuse_a=*/false, /*reuse_b=*/false);
    }

    __shared__ float smem[32];
    smem[lane] = c[0];               // lanes 0..15: M=0, N=lane
    __syncthreads();
    if (lane == 0) {
        float sum = 0.0f;
        for (int n = 0; n < 16; ++n) sum += smem[n];
        out[0] = sum * invN;
    }
}

extern "C" void kernel_launch(void* const* d_in, const int* in_sizes, int n_in,
                              void* d_out, int out_size, void* d_ws, size_t ws_size,
                              hipStream_t stream) {
    const float* pred = (const float*)d_in[0];
    const float* targ = (const float*)d_in[1];
    float* out        = (float*)d_out;
    float* partials   = (float*)d_ws;

    const int n  = in_sizes[0];      // 19,660,800
    const int n4 = n / 4;

    // Partial count: multiple of 64 (WMMA folds 64/iter), clamped to scratch.
    int blocks = 2048;
    const int maxPart = (int)(ws_size / sizeof(float));
    if (blocks > maxPart) blocks = (maxPart / 64) * 64;
    if (blocks < 64) blocks = 64;

    const int threads = 256;         // 8 wave32 per block

    svl_partial_kernel<<<blocks, threads, 0, stream>>>(pred, targ, partials, n4, n);
    svl_final_wmma_kernel<<<1, 32, 0, stream>>>(partials, out, blocks, 1.0f / (float)n);
}